// Layer_4140348473412
// MI455X (gfx1250) — compile-verified
//
#include <hip/hip_runtime.h>
#include <hip/hip_bf16.h>
#include <math.h>

typedef __attribute__((ext_vector_type(2))) float v2f;
typedef __attribute__((ext_vector_type(4))) float v4f;
typedef __attribute__((ext_vector_type(8))) float v8f;

#define BATCH 1024
#define DIM   256
#define EPS   1e-6f

// ---------------------------------------------------------------------------
// GEMM + bias + activation:  out[m][n] = act( sum_k A[m][k]*W[n][k] + bias[n] )
// A: [1024,256], W: [256,256] (row-major, torch Linear weight), out: [1024,256]
// One wave32 computes one 16x16 tile via V_WMMA_F32_16X16X4_F32 (K stepped by 4).
// blockDim = (32,4): 4 waves -> 4 adjacent N-tiles. EXEC all-ones (WMMA req).
// act: 0 = sigmoid, 1 = tanh
// ---------------------------------------------------------------------------
__global__ __launch_bounds__(128) void gemm_bias_act(
    const float* __restrict__ A,
    const float* __restrict__ W,
    const float* __restrict__ bias,
    float* __restrict__ out,
    int act)
{
    const int lane = threadIdx.x;          // 0..31
    const int half = lane >> 4;            // 0 or 1
    const int lm   = lane & 15;

    const int m0 = blockIdx.x * 16;
    const int n0 = (blockIdx.y * 4 + threadIdx.y) * 16;

    // A fragment (16x4, f32): VGPR0 holds K=k (lanes 0-15) / K=k+2 (lanes 16-31),
    // VGPR1 holds K=k+1 / K=k+3  ->  per-lane float2 at column (k + 2*half).
    const float* Arow = A + (size_t)(m0 + lm) * DIM + 2 * half;
    // B fragment (4x16) with B[k][n] = W[n][k]: mirrored layout.
    const float* Wrow = W + (size_t)(n0 + lm) * DIM + 2 * half;

    v8f acc = {};
#pragma unroll 8
    for (int k = 0; k < DIM; k += 4) {
        v2f a = *(const v2f*)(Arow + k);
        v2f b = *(const v2f*)(Wrow + k);
        // 8 args: (neg_a, A, neg_b, B, c_mod, C, reuse_a, reuse_b)
        acc = __builtin_amdgcn_wmma_f32_16x16x4_f32(
            false, a, false, b, (short)0, acc, false, false);
    }

    const float bn = bias[n0 + lm];
#pragma unroll
    for (int r = 0; r < 8; ++r) {
        const int row = m0 + r + 8 * half;   // C/D layout: VGPR r -> M=r (+8 for hi lanes)
        float v = acc[r] + bn;
        v = (act == 0) ? (1.0f / (1.0f + expf(-v))) : tanhf(v);
        out[(size_t)row * DIM + (n0 + lm)] = v;
    }
}

// ---------------------------------------------------------------------------
// Per-batch scalars + row/col vectors + z.
// One 256-thread block per batch row b.
//   q_j = sqrt(1/d_j + eps)
//   S   = sum u ; s = sum u^2/d + eps*S^2 ; right = (1 - sqrt(1/(1+s)))/s
//   t_j = sqrt(eps)*S + u_j*(q_j - sqrt(eps))
//   a_i = right*(u_i/d_i + eps*S)
//   z_i = sqrt(eps)*sum(v) + (q_i - sqrt(eps))*v_i - a_i*sum(t_j v_j) + mean_i
// ---------------------------------------------------------------------------
__device__ __forceinline__ float block_sum256(float val, float* red) {
    const int t = threadIdx.x;
    red[t] = val;
    __syncthreads();
#pragma unroll
    for (int s = 128; s > 0; s >>= 1) {
        if (t < s) red[t] += red[t + s];
        __syncthreads();
    }
    const float r = red[0];
    __syncthreads();
    return r;
}

__global__ __launch_bounds__(256) void finalize_stats(
    const float* __restrict__ d, const float* __restrict__ u,
    const float* __restrict__ v, const float* __restrict__ mean,
    float* __restrict__ t_arr, float* __restrict__ a_arr,
    float* __restrict__ q_arr, float* __restrict__ z)
{
    __shared__ float red[256];
    const int b = blockIdx.x;
    const int j = threadIdx.x;
    const size_t idx = (size_t)b * DIM + j;

    const float dj = d[idx];
    const float uj = u[idx];
    const float vj = v[idx];
    const float qj = sqrtf(1.0f / dj + EPS);
    const float se = sqrtf(EPS);

    const float S   = block_sum256(uj, red);
    const float s2  = block_sum256(uj * uj / dj, red);
    const float Sv  = block_sum256(vj, red);

    const float s     = s2 + EPS * S * S;
    const float eta   = 1.0f / (1.0f + s);
    const float right = (1.0f - sqrtf(eta)) / s;

    const float tj = se * S + uj * (qj - se);
    const float aj = right * (uj / dj + EPS * S);

    const float Stv = block_sum256(tj * vj, red);

    // z is write-only output -> non-temporal store
    __builtin_nontemporal_store(se * Sv + (qj - se) * vj - aj * Stv + mean[idx],
                                z + idx);
    t_arr[idx] = tj;
    a_arr[idx] = aj;
    q_arr[idx] = qj;
}

// ---------------------------------------------------------------------------
// R writer (bandwidth-bound: 268 MB of stores -> HBM roofline ~11.5us).
// R[b][i][j] = (i==j ? q_i : sqrt(eps)) - a_i * t_j
// R is written once and never re-read, and exceeds the 192MB L2 -> use
// non-temporal 128-bit stores so we don't thrash L2 / write-allocate.
// grid (64, 1024), block 256: 4 rows per block, float4 stores, coalesced rows.
// ---------------------------------------------------------------------------
__global__ __launch_bounds__(256) void write_R(
    const float* __restrict__ t_arr, const float* __restrict__ a_arr,
    const float* __restrict__ q_arr, float* __restrict__ R)
{
    const int b  = blockIdx.y;
    const int i  = blockIdx.x * 4 + (threadIdx.x >> 6);
    const int j0 = (threadIdx.x & 63) * 4;

    const size_t base = (size_t)b * DIM;
    const float  ai = a_arr[base + i];
    const float  qi = q_arr[base + i];
    const float  se = sqrtf(EPS);

    const v4f t = *(const v4f*)(t_arr + base + j0);
    v4f r;
    r.x = ((i == j0 + 0) ? qi : se) - ai * t.x;
    r.y = ((i == j0 + 1) ? qi : se) - ai * t.y;
    r.z = ((i == j0 + 2) ? qi : se) - ai * t.z;
    r.w = ((i == j0 + 3) ? qi : se) - ai * t.w;

    __builtin_nontemporal_store(r, (v4f*)(R + (base + i) * DIM + j0));
}

// ---------------------------------------------------------------------------
extern "C" void kernel_launch(void* const* d_in, const int* in_sizes, int n_in,
                              void* d_out, int out_size, void* d_ws, size_t ws_size,
                              hipStream_t stream) {
    (void)in_sizes; (void)n_in; (void)out_size; (void)ws_size;

    const float* x   = (const float*)d_in[0];
    const float* Wd1 = (const float*)d_in[1];  const float* bd1 = (const float*)d_in[2];
    const float* Wd2 = (const float*)d_in[3];  const float* bd2 = (const float*)d_in[4];
    const float* Wu1 = (const float*)d_in[5];  const float* bu1 = (const float*)d_in[6];
    const float* Wu2 = (const float*)d_in[7];  const float* bu2 = (const float*)d_in[8];
    const float* Wm1 = (const float*)d_in[9];  const float* bm1 = (const float*)d_in[10];
    const float* Wm2 = (const float*)d_in[11]; const float* bm2 = (const float*)d_in[12];
    const float* v   = (const float*)d_in[13];

    // Output layout: mean [1024*256] | R [1024*256*256] | z [1024*256]
    float* out_mean = (float*)d_out;
    float* out_R    = out_mean + (size_t)BATCH * DIM;
    float* out_z    = out_R + (size_t)BATCH * DIM * DIM;

    // Workspace: h1 | d | u | t | a | q  (each 1024*256 f32 = 1 MB)
    const size_t n = (size_t)BATCH * DIM;
    float* h1    = (float*)d_ws;
    float* d_buf = h1 + n;
    float* u_buf = d_buf + n;
    float* t_arr = u_buf + n;
    float* a_arr = t_arr + n;
    float* q_arr = a_arr + n;

    const dim3 gGrid(BATCH / 16, (DIM / 16) / 4);   // (64, 4)
    const dim3 gBlk(32, 4);

    // d-path (sigmoid, sigmoid)
    gemm_bias_act<<<gGrid, gBlk, 0, stream>>>(x,  Wd1, bd1, h1,    0);
    gemm_bias_act<<<gGrid, gBlk, 0, stream>>>(h1, Wd2, bd2, d_buf, 0);
    // u-path (sigmoid, sigmoid)
    gemm_bias_act<<<gGrid, gBlk, 0, stream>>>(x,  Wu1, bu1, h1,    0);
    gemm_bias_act<<<gGrid, gBlk, 0, stream>>>(h1, Wu2, bu2, u_buf, 0);
    // mean-path (tanh, tanh) -> straight into d_out
    gemm_bias_act<<<gGrid, gBlk, 0, stream>>>(x,  Wm1, bm1, h1,       1);
    gemm_bias_act<<<gGrid, gBlk, 0, stream>>>(h1, Wm2, bm2, out_mean, 1);

    finalize_stats<<<BATCH, 256, 0, stream>>>(d_buf, u_buf, v, out_mean,
                                              t_arr, a_arr, q_arr, out_z);

    write_R<<<dim3(DIM / 4, BATCH), 256, 0, stream>>>(t_arr, a_arr, q_arr, out_R);
}